// LinearAttention_63625645523107
// MI455X (gfx1250) — compile-verified
//
#include <hip/hip_runtime.h>
#include <math.h>

// ---------------------------------------------------------------------------
// LinearAttention on gfx1250 (MI455X), full fp32 pipeline using
// V_WMMA_F32_16X16X4_F32 for every contraction, async global->LDS staging
// for the (non-transposed) A tiles.
//
// Pipeline:
//   1) qkv[b]  = w_qkv @ x[b]                       (WMMA GEMM, 128x128 tiles)
//   2) k rows  = softmax over n (rows 512..1023)    (block reduction)
//   3) ctx[b,h]= k_sm[64x4096] @ v[64x4096]^T       (WMMA + chunked atomics)
//   4) W2[b][:,h*64+d] = w_out[:,h*64+e] ctx[d,e]   (WMMA)
//   5) out[b]  = W2[b] @ q[b] + b_out               (WMMA GEMM, 128x128 tiles)
//
// Workspace layout in d_ws (~201 MiB):
//   qkv : 8*1536*4096 f32   (192 MiB)
//   ctx : 8*8*64*64  f32    (  1 MiB)
//   W2  : 8*512*512  f32    (  8 MiB)
// ---------------------------------------------------------------------------

typedef float v2f __attribute__((ext_vector_type(2)));
typedef float v8f __attribute__((ext_vector_type(8)));

constexpr int BATCH  = 8;
constexpr int DIM    = 512;
constexpr int HEADS  = 8;
constexpr int DH     = 64;
constexpr int NTOK   = 4096;   // 64*64 spatial
constexpr int O3     = 1536;   // 3*HIDDEN

__device__ __forceinline__ v8f wmma_f32(v2f a, v2f b, v8f c) {
  // D = A(16x4) * B(4x16) + C(16x16), fp32
  return __builtin_amdgcn_wmma_f32_16x16x4_f32(
      /*neg_a=*/false, a, /*neg_b=*/false, b,
      /*c_mod=*/(short)0, c, /*reuse_a=*/false, /*reuse_b=*/false);
}

// Async copy of 16 bytes global -> LDS (per lane), tracked by ASYNCcnt.
__device__ __forceinline__ void async_g2l_b128(unsigned lds_off, const float* gaddr) {
  asm volatile("global_load_async_to_lds_b128 %0, %1, off"
               :: "v"(lds_off), "v"(gaddr) : "memory");
}
__device__ __forceinline__ void wait_async0() {
  asm volatile("s_wait_asynccnt 0x0" ::: "memory");
}

// ---------------------------------------------------------------------------
// WMMA GEMM: C[b] = A[b][M x K] * B[b][K x 4096] (+bias), row-major.
// grid = (4096/128, M/128, batch), block = 256 (8 waves).
// Block tile 128x128, wave tile 32x64 (2 M-subtiles x 4 N-subtiles,
// 8 f32 accumulators). LDS double-buffered; A staged with async
// global->LDS copies (no VGPR round-trip), B staged transposed via VGPRs so
// every WMMA operand pair is one aligned ds_load_b64.
// ---------------------------------------------------------------------------
__global__ __launch_bounds__(256)
void gemm128x128(const float* __restrict__ A, long long aBatchStride, int lda,
                 const float* __restrict__ B, long long bBatchStride,
                 float* __restrict__ C, long long cBatchStride,
                 int K, const float* __restrict__ bias)
{
  __shared__ __align__(16) float As [2][128][20];  // [buf][m][k]  (pad 20: 16B rows)
  __shared__ __align__(16) float Bst[2][128][18];  // [buf][n][k]  (pad 18: b64 pairs)

  const int b  = blockIdx.z;
  const float* Ab = A + (size_t)b * aBatchStride;
  const float* Bb = B + (size_t)b * bBatchStride;
  float*       Cb = C + (size_t)b * cBatchStride;

  const int n0 = blockIdx.x * 128;
  const int m0 = blockIdx.y * 128;

  const int t     = threadIdx.x;
  const int lane  = t & 31;
  const int w     = t >> 5;
  const int wm    = w >> 1;        // 0..3 : 32-row M quadrant
  const int wn    = w & 1;         // 0..1 : 64-col N half
  const int lrow  = lane & 15;     // M row (A) / N col (B)
  const int khalf = lane >> 4;     // selects K pair {0,1} vs {2,3}

  // cooperative staging maps
  const int ar  = t >> 1;          // 0..127   (A row)
  const int ac  = (t & 1) * 8;     // 0 or 8   (A col base, 8 floats)
  const int br2 = t >> 5;          // 0..7     (B k-row pair)
  const int bc2 = (t & 31) * 4;    // 0..124   (B col base, 4 floats)

  v8f acc[2][4];
#pragma unroll
  for (int i = 0; i < 2; ++i)
#pragma unroll
    for (int j = 0; j < 4; ++j) acc[i][j] = {};

  float4 rb0, rb1;

  // A tile: two async b128 copies per thread, straight into padded LDS rows.
  auto gloadA_async = [&](int k0, int buf) {
    const float* ap = Ab + (size_t)(m0 + ar) * lda + k0 + ac;
    const unsigned l0 = (unsigned)(uintptr_t)(&As[buf][ar][ac]);
    async_g2l_b128(l0,      ap);
    async_g2l_b128(l0 + 16, ap + 4);
  };

  auto gloadB = [&](int k0) {
    const float* bp = Bb + (size_t)(k0 + 2 * br2) * NTOK + n0 + bc2;
    rb0 = *reinterpret_cast<const float4*>(bp);
    rb1 = *reinterpret_cast<const float4*>(bp + NTOK);
  };

  auto sstoreB = [&](int buf) {
    // transpose B: Bst[n][k] ; write (k, k+1) pairs as b64
    v2f p0 = {rb0.x, rb1.x}; *reinterpret_cast<v2f*>(&Bst[buf][bc2 + 0][2 * br2]) = p0;
    v2f p1 = {rb0.y, rb1.y}; *reinterpret_cast<v2f*>(&Bst[buf][bc2 + 1][2 * br2]) = p1;
    v2f p2 = {rb0.z, rb1.z}; *reinterpret_cast<v2f*>(&Bst[buf][bc2 + 2][2 * br2]) = p2;
    v2f p3 = {rb0.w, rb1.w}; *reinterpret_cast<v2f*>(&Bst[buf][bc2 + 3][2 * br2]) = p3;
  };

  auto compute = [&](int buf) {
#pragma unroll
    for (int kb = 0; kb < 16; kb += 4) {
      const int kk = kb + 2 * khalf;
      v2f a0 = *reinterpret_cast<const v2f*>(&As[buf][wm * 32 + lrow][kk]);
      v2f a1 = *reinterpret_cast<const v2f*>(&As[buf][wm * 32 + 16 + lrow][kk]);
#pragma unroll
      for (int nj = 0; nj < 4; ++nj) {
        v2f bv = *reinterpret_cast<const v2f*>(&Bst[buf][wn * 64 + nj * 16 + lrow][kk]);
        acc[0][nj] = wmma_f32(a0, bv, acc[0][nj]);
        acc[1][nj] = wmma_f32(a1, bv, acc[1][nj]);
      }
    }
  };

  const int nt = K / 16;
  gloadA_async(0, 0);
  gloadB(0);
  sstoreB(0);
  wait_async0();
  __syncthreads();
  for (int it = 0; it < nt; ++it) {
    const int cur = it & 1;
    if (it + 1 < nt) {
      gloadA_async((it + 1) * 16, cur ^ 1);  // async into other buffer
      gloadB((it + 1) * 16);                 // overlaps compute on `cur`
    }
    compute(cur);
    if (it + 1 < nt) {
      sstoreB(cur ^ 1);                      // other buffer: safe (barrier @ it-1)
      wait_async0();
      __syncthreads();
    }
  }

  // C/D layout: VGPR r -> M = r + 8*khalf, N = lane&15
#pragma unroll
  for (int mi = 0; mi < 2; ++mi) {
    const int mrow = m0 + wm * 32 + mi * 16 + khalf * 8;
#pragma unroll
    for (int nj = 0; nj < 4; ++nj) {
      const int nc = n0 + wn * 64 + nj * 16 + lrow;
#pragma unroll
      for (int r = 0; r < 8; ++r) {
        const float bi = bias ? bias[mrow + r] : 0.0f;
        Cb[(size_t)(mrow + r) * NTOK + nc] = acc[mi][nj][r] + bi;
      }
    }
  }
}

// ---------------------------------------------------------------------------
// Row softmax over n=4096 for the k block (rows 512..1023 of qkv[b]).
// grid = (512, batch), block = 256; each thread owns 16 elements.
// ---------------------------------------------------------------------------
__global__ void softmax_k(float* __restrict__ qkv)
{
  float* row = qkv + ((size_t)blockIdx.y * O3 + DIM + blockIdx.x) * NTOK;
  __shared__ float red[256];
  const int t = threadIdx.x;

  float vals[16];
  float vmax = -INFINITY;
#pragma unroll
  for (int i = 0; i < 16; ++i) {
    vals[i] = row[t + i * 256];
    vmax = fmaxf(vmax, vals[i]);
  }
  red[t] = vmax; __syncthreads();
  for (int s = 128; s > 0; s >>= 1) {
    if (t < s) red[t] = fmaxf(red[t], red[t + s]);
    __syncthreads();
  }
  vmax = red[0]; __syncthreads();

  float sum = 0.0f;
#pragma unroll
  for (int i = 0; i < 16; ++i) {
    vals[i] = __expf(vals[i] - vmax);
    sum += vals[i];
  }
  red[t] = sum; __syncthreads();
  for (int s = 128; s > 0; s >>= 1) {
    if (t < s) red[t] += red[t + s];
    __syncthreads();
  }
  const float inv = 1.0f / red[0];
#pragma unroll
  for (int i = 0; i < 16; ++i) row[t + i * 256] = vals[i] * inv;
}

// ---------------------------------------------------------------------------
// ctx[b,h][d,e] = sum_n k_sm[d,n] * v[e,n]   (64x64 output, K = 4096)
// grid = (8 n-chunks of 512, HEADS, batch); partial sums via f32 atomics.
// ---------------------------------------------------------------------------
__global__ void ctx_partial(const float* __restrict__ qkv, float* __restrict__ ctx)
{
  __shared__ __align__(16) float Ks[64][18];   // [d][n'] (pad 18: b64 pairs)
  __shared__ __align__(16) float Vt[16][65];   // [n'][e]

  const int chunk = blockIdx.x;
  const int h     = blockIdx.y;
  const int b     = blockIdx.z;

  const float* kp = qkv + ((size_t)b * O3 + DIM     + h * DH) * NTOK;  // k_sm rows
  const float* vp = qkv + ((size_t)b * O3 + 2 * DIM + h * DH) * NTOK;  // v rows

  const int t     = threadIdx.x;
  const int lane  = t & 31;
  const int w     = t >> 5;
  const int wr    = w >> 1;
  const int wc    = w & 1;
  const int lrow  = lane & 15;
  const int khalf = lane >> 4;

  const int krow = t >> 2;          // d   : 0..63
  const int kcol = (t & 3) * 4;     // n'  : 0..12

  v8f acc0 = {}; v8f acc1 = {};

  const int nbase = chunk * 512;
  for (int nb = nbase; nb < nbase + 512; nb += 16) {
    const float4 kv = *reinterpret_cast<const float4*>(kp + (size_t)krow * NTOK + nb + kcol);
    const float4 vv = *reinterpret_cast<const float4*>(vp + (size_t)krow * NTOK + nb + kcol);
    __syncthreads();
    Ks[krow][kcol + 0] = kv.x; Ks[krow][kcol + 1] = kv.y;
    Ks[krow][kcol + 2] = kv.z; Ks[krow][kcol + 3] = kv.w;
    Vt[kcol + 0][krow] = vv.x; Vt[kcol + 1][krow] = vv.y;
    Vt[kcol + 2][krow] = vv.z; Vt[kcol + 3][krow] = vv.w;
    __syncthreads();

#pragma unroll
    for (int kb = 0; kb < 16; kb += 4) {
      const int kk = kb + 2 * khalf;
      v2f a = *reinterpret_cast<const v2f*>(&Ks[wr * 16 + lrow][kk]);
      v2f b0, b1;
      b0.x = Vt[kk + 0][wc * 32 + lrow];
      b0.y = Vt[kk + 1][wc * 32 + lrow];
      b1.x = Vt[kk + 0][wc * 32 + 16 + lrow];
      b1.y = Vt[kk + 1][wc * 32 + 16 + lrow];
      acc0 = wmma_f32(a, b0, acc0);
      acc1 = wmma_f32(a, b1, acc1);
    }
  }

  float* cb = ctx + ((size_t)(b * HEADS + h)) * DH * DH;
  const int drow = wr * 16 + khalf * 8;
  const int ec0  = wc * 32 + lrow;
#pragma unroll
  for (int r = 0; r < 8; ++r) {
    atomicAdd(&cb[(drow + r) * DH + ec0],      acc0[r]);
    atomicAdd(&cb[(drow + r) * DH + ec0 + 16], acc1[r]);
  }
}

// ---------------------------------------------------------------------------
// W2[b][o, h*64+d] = sum_e w_out[o, h*64+e] * ctx[b,h][d,e]
// grid = (8 o-groups of 64, HEADS, batch), block = 256.
// ---------------------------------------------------------------------------
__global__ void w2_kernel(const float* __restrict__ w_out,
                          const float* __restrict__ ctx,
                          float* __restrict__ W2)
{
  __shared__ __align__(16) float Cs[64][66];   // Cs[d][e], pad 66: b64 pairs

  const int og = blockIdx.x;     // o-group
  const int h  = blockIdx.y;
  const int b  = blockIdx.z;
  const int o0 = og * 64;

  const float* cb = ctx + ((size_t)(b * HEADS + h)) * DH * DH;
  for (int i = threadIdx.x; i < DH * DH; i += 256)
    Cs[i >> 6][i & 63] = cb[i];
  __syncthreads();

  const int t     = threadIdx.x;
  const int lane  = t & 31;
  const int w     = t >> 5;
  const int wr    = w >> 1;
  const int wc    = w & 1;
  const int lrow  = lane & 15;
  const int khalf = lane >> 4;

  v8f acc0 = {}; v8f acc1 = {};

  // A[o,e] = w_out[o*512 + h*64 + e] ; B[e,d] = ctx[d,e] = Cs[d][e]
#pragma unroll
  for (int kb = 0; kb < DH; kb += 4) {
    const int kk = kb + 2 * khalf;
    v2f a, b0, b1;
    const float* arow = w_out + (size_t)(o0 + lrow) * DIM + h * DH;
    a.x = arow[kk + 0];
    a.y = arow[kk + 1];
    b0 = *reinterpret_cast<const v2f*>(&Cs[wc * 32 + lrow][kk]);
    b1 = *reinterpret_cast<const v2f*>(&Cs[wc * 32 + 16 + lrow][kk]);
    acc0 = wmma_f32(a, b0, acc0);
    acc1 = wmma_f32(a, b1, acc1);
  }

  float* w2b = W2 + (size_t)b * DIM * DIM;
  const int orow = o0 + wr * 16 + khalf * 8;
  const int dc0  = wc * 32 + lrow;
#pragma unroll
  for (int r = 0; r < 8; ++r) {
    w2b[(size_t)(orow + r) * DIM + h * DH + dc0]      = acc0[r];
    w2b[(size_t)(orow + r) * DIM + h * DH + dc0 + 16] = acc1[r];
  }
}

// ---------------------------------------------------------------------------
extern "C" void kernel_launch(void* const* d_in, const int* in_sizes, int n_in,
                              void* d_out, int out_size, void* d_ws, size_t ws_size,
                              hipStream_t stream) {
  (void)in_sizes; (void)n_in; (void)out_size; (void)ws_size;
  const float* x     = (const float*)d_in[0];   // (8,512,64,64)
  const float* w_qkv = (const float*)d_in[1];   // (1536,512)
  const float* w_out = (const float*)d_in[2];   // (512,512)
  const float* b_out = (const float*)d_in[3];   // (512,)
  float* out = (float*)d_out;

  float* qkv = (float*)d_ws;                                  // 8*1536*4096
  float* ctx = qkv + (size_t)BATCH * O3 * NTOK;               // 8*8*64*64
  float* W2  = ctx + (size_t)BATCH * HEADS * DH * DH;         // 8*512*512

  // 1) qkv = w_qkv @ x[b]
  gemm128x128<<<dim3(NTOK / 128, O3 / 128, BATCH), 256, 0, stream>>>(
      w_qkv, 0LL, DIM,
      x,   (long long)DIM * NTOK,
      qkv, (long long)O3 * NTOK,
      DIM, nullptr);

  // 2) softmax over n on k rows
  softmax_k<<<dim3(DIM, BATCH), 256, 0, stream>>>(qkv);

  // 3) ctx = k_sm @ v^T  (chunked + atomics; zero first, ws is poisoned)
  hipMemsetAsync(ctx, 0, (size_t)BATCH * HEADS * DH * DH * sizeof(float), stream);
  ctx_partial<<<dim3(8, HEADS, BATCH), 256, 0, stream>>>(qkv, ctx);

  // 4) W2[b] = w_out folded with ctx^T
  w2_kernel<<<dim3(DIM / 64, HEADS, BATCH), 256, 0, stream>>>(w_out, ctx, W2);

  // 5) out = W2[b] @ q[b] + b_out   (q = first 512 rows of qkv[b])
  gemm128x128<<<dim3(NTOK / 128, DIM / 128, BATCH), 256, 0, stream>>>(
      W2,  (long long)DIM * DIM, DIM,
      qkv, (long long)O3 * NTOK,
      out, (long long)DIM * NTOK,
      DIM, b_out);
}